// LinearCrossAttention_65687229825324
// MI455X (gfx1250) — compile-verified
//
#include <hip/hip_runtime.h>
#include <math.h>

// Problem constants (fixed by the reference)
#define B_     8
#define M_     4096
#define N_     4096
#define DM     1024   // d_model
#define DC     768    // d_ctx
#define DH     16     // d_head

typedef float v2f __attribute__((ext_vector_type(2)));
typedef float v8f __attribute__((ext_vector_type(8)));

__device__ __forceinline__ v8f wmma_f32(v2f a, v2f b, v8f c) {
    // D = A(16x4 f32) * B(4x16 f32) + C(16x16 f32)
    return __builtin_amdgcn_wmma_f32_16x16x4_f32(
        /*neg_a=*/false, a, /*neg_b=*/false, b,
        /*c_mod=*/(short)0, c, /*reuse_a=*/false, /*reuse_b=*/false);
}

__device__ __forceinline__ float elu1(float x) {
    // elu(x)+1 : x>0 -> x+1 ; x<=0 -> exp(x)
    return x > 0.0f ? x + 1.0f : __expf(x);
}

// ---------------------------------------------------------------------------
// Kernel 0: zero the workspace accumulators (KtV: B*256 floats, Ksum: B*16)
// ---------------------------------------------------------------------------
__global__ void lca_zero_ws(float* __restrict__ p, int n) {
    int i = blockIdx.x * blockDim.x + threadIdx.x;
    if (i < n) p[i] = 0.0f;
}

// ---------------------------------------------------------------------------
// Kernel 1: K = elu1(ctx@Wk+bk), V = ctx@Wv+bv  (never materialized),
//           reduce KtV (16x16 per batch) and Ksum (16 per batch).
// grid = (N/128, B), block = 128; each wave owns TWO 16-row tiles so the
// Wk/Wv B-fragments are shared across two A-tiles (4 WMMA per weight load set)
// ---------------------------------------------------------------------------
__global__ void __launch_bounds__(128)
lca_kv_reduce(const float* __restrict__ ctx,
              const float* __restrict__ Wk, const float* __restrict__ bk,
              const float* __restrict__ Wv, const float* __restrict__ bv,
              float* __restrict__ ktv, float* __restrict__ ksum)
{
    const int b    = blockIdx.y;
    const int wave = threadIdx.x >> 5;
    const int lane = threadIdx.x & 31;
    const int hi   = lane >> 4;      // 0 or 1
    const int lo   = lane & 15;      // row (A) / col (B,C,D)

    const int row0 = (blockIdx.x * 8 + wave * 2) * 16;       // 2 tiles: row0, row0+16
    const float* cb0 = ctx + ((size_t)b * N_ + row0) * DC;
    const float* cb1 = cb0 + (size_t)16 * DC;

    v8f kc0 = {}, kc1 = {}, vc0 = {}, vc1 = {};
    #pragma unroll 4
    for (int k0 = 0; k0 < DC; k0 += 4) {
        const int kk = k0 + 2 * hi;
        if ((k0 & 28) == 0) {  // once per 8 k-steps: prefetch next 128B line of A stream
            __builtin_prefetch(cb0 + (size_t)lo * DC + k0 + 32, 0, 0);
            __builtin_prefetch(cb1 + (size_t)lo * DC + k0 + 32, 0, 0);
        }
        v2f a0 = *(const v2f*)(cb0 + (size_t)lo * DC + kk);
        v2f a1 = *(const v2f*)(cb1 + (size_t)lo * DC + kk);
        v2f wk2; wk2.x = Wk[kk * DH + lo]; wk2.y = Wk[(kk + 1) * DH + lo];
        v2f wv2; wv2.x = Wv[kk * DH + lo]; wv2.y = Wv[(kk + 1) * DH + lo];
        kc0 = wmma_f32(a0, wk2, kc0);
        kc1 = wmma_f32(a1, wk2, kc1);
        vc0 = wmma_f32(a0, wv2, vc0);
        vc1 = wmma_f32(a1, wv2, vc1);
    }

    __shared__ float Kl[8][16][16];
    __shared__ float Vl[8][16][16];
    const float bkl = bk[lo];
    const float bvl = bv[lo];
    const int t0 = wave * 2, t1 = t0 + 1;
    #pragma unroll
    for (int j = 0; j < 8; ++j) {
        const int r = j + 8 * hi;               // C/D layout: row = vgpr + 8*hi
        Kl[t0][r][lo] = elu1(kc0[j] + bkl);
        Kl[t1][r][lo] = elu1(kc1[j] + bkl);
        Vl[t0][r][lo] = vc0[j] + bvl;
        Vl[t1][r][lo] = vc1[j] + bvl;
    }
    __syncthreads();

    // Block-partial KtV: sum over this block's 128 context rows
    for (int p = threadIdx.x; p < 256; p += 128) {
        const int d = p >> 4, e = p & 15;
        float s = 0.0f;
        #pragma unroll 4
        for (int r = 0; r < 128; ++r)
            s += Kl[r >> 4][r & 15][d] * Vl[r >> 4][r & 15][e];
        atomicAdd(&ktv[b * 256 + p], s);
    }
    if (threadIdx.x < 16) {
        float s = 0.0f;
        #pragma unroll 4
        for (int r = 0; r < 128; ++r)
            s += Kl[r >> 4][r & 15][threadIdx.x];
        atomicAdd(&ksum[b * DH + threadIdx.x], s);
    }
}

// ---------------------------------------------------------------------------
// Kernel 2: Q = elu1(x@Wq+bq); num = Q@KtV; denom = Q.Ksum + 1e-6;
//           out = (num/denom)@Wo + bo
// grid = (M/128, B), block = 128; TWO 16-row M-tiles per wave
// ---------------------------------------------------------------------------
__global__ void __launch_bounds__(128)
lca_q_out(const float* __restrict__ x,
          const float* __restrict__ Wq, const float* __restrict__ bq,
          const float* __restrict__ ktv, const float* __restrict__ ksum,
          const float* __restrict__ Wo, const float* __restrict__ bo,
          float* __restrict__ out)
{
    const int b    = blockIdx.y;
    const int wave = threadIdx.x >> 5;
    const int lane = threadIdx.x & 31;
    const int hi   = lane >> 4;
    const int lo   = lane & 15;

    const int m0 = (blockIdx.x * 8 + wave * 2) * 16;         // 2 tiles: m0, m0+16
    const float* xb0 = x + ((size_t)b * M_ + m0) * DM;
    const float* xb1 = xb0 + (size_t)16 * DM;

    // ---- Q projection: 256 fp32-WMMA k-steps over d_model, 2 tiles ----
    v8f qc0 = {}, qc1 = {};
    #pragma unroll 4
    for (int k0 = 0; k0 < DM; k0 += 4) {
        const int kk = k0 + 2 * hi;
        if ((k0 & 28) == 0) {
            __builtin_prefetch(xb0 + (size_t)lo * DM + k0 + 32, 0, 0);
            __builtin_prefetch(xb1 + (size_t)lo * DM + k0 + 32, 0, 0);
        }
        v2f a0 = *(const v2f*)(xb0 + (size_t)lo * DM + kk);
        v2f a1 = *(const v2f*)(xb1 + (size_t)lo * DM + kk);
        v2f w; w.x = Wq[kk * DH + lo]; w.y = Wq[(kk + 1) * DH + lo];
        qc0 = wmma_f32(a0, w, qc0);
        qc1 = wmma_f32(a1, w, qc1);
    }

    __shared__ float Tl[8][16][16];
    const int t0 = wave * 2, t1 = t0 + 1;
    const float bql = bq[lo];
    const float ks  = ksum[b * DH + lo];
    float denom0[8], denom1[8];
    #pragma unroll
    for (int j = 0; j < 8; ++j) {
        float q0 = elu1(qc0[j] + bql);
        float q1 = elu1(qc1[j] + bql);
        Tl[t0][j + 8 * hi][lo] = q0;      // stage Q (C/D layout -> LDS row-major)
        Tl[t1][j + 8 * hi][lo] = q1;
        denom0[j] = q0 * ks;              // partial Q[m][lo] * Ksum[lo]
        denom1[j] = q1 * ks;
    }
    // reduce denom over the 16 lanes that share a row group
    #pragma unroll
    for (int mask = 1; mask < 16; mask <<= 1) {
        #pragma unroll
        for (int j = 0; j < 8; ++j) {
            denom0[j] += __shfl_xor(denom0[j], mask, 32);
            denom1[j] += __shfl_xor(denom1[j], mask, 32);
        }
    }
    __syncthreads();

    // ---- num = Q @ KtV (contraction over d_head, 4 k-steps, shared B) ----
    const float* Kt = ktv + b * 256;
    v2f kb[4];
    #pragma unroll
    for (int s = 0; s < 4; ++s) {
        const int kk = 4 * s + 2 * hi;
        kb[s].x = Kt[kk * DH + lo];
        kb[s].y = Kt[(kk + 1) * DH + lo];
    }
    v8f num0 = {}, num1 = {};
    #pragma unroll
    for (int s = 0; s < 4; ++s) {
        const int kk = 4 * s + 2 * hi;
        v2f a0 = *(const v2f*)(&Tl[t0][lo][kk]);     // Q A-fragments from LDS
        v2f a1 = *(const v2f*)(&Tl[t1][lo][kk]);
        num0 = wmma_f32(a0, kb[s], num0);
        num1 = wmma_f32(a1, kb[s], num1);
    }
    __syncthreads();

    // ---- y = num / denom, restage for out-projection ----
    #pragma unroll
    for (int j = 0; j < 8; ++j) {
        Tl[t0][j + 8 * hi][lo] = num0[j] / (denom0[j] + 1e-6f);
        Tl[t1][j + 8 * hi][lo] = num1[j] / (denom1[j] + 1e-6f);
    }
    __syncthreads();

    v2f ya0[4], ya1[4];
    #pragma unroll
    for (int s = 0; s < 4; ++s) {
        const int kk = 4 * s + 2 * hi;
        ya0[s] = *(const v2f*)(&Tl[t0][lo][kk]);
        ya1[s] = *(const v2f*)(&Tl[t1][lo][kk]);
    }

    // ---- out = y @ Wo + bo : 64 column tiles of 16, Wo frags shared ----
    float* ob0 = out + ((size_t)b * M_ + m0) * DM;
    float* ob1 = ob0 + (size_t)16 * DM;
    for (int ct = 0; ct < 64; ++ct) {
        const int col = ct * 16 + lo;
        v8f acc0 = {}, acc1 = {};
        #pragma unroll
        for (int s = 0; s < 4; ++s) {
            const int kk = 4 * s + 2 * hi;
            v2f bb; bb.x = Wo[kk * DM + col]; bb.y = Wo[(kk + 1) * DM + col];
            acc0 = wmma_f32(ya0[s], bb, acc0);
            acc1 = wmma_f32(ya1[s], bb, acc1);
        }
        const float bol = bo[col];
        #pragma unroll
        for (int j = 0; j < 8; ++j) {
            ob0[(size_t)(j + 8 * hi) * DM + col] = acc0[j] + bol;
            ob1[(size_t)(j + 8 * hi) * DM + col] = acc1[j] + bol;
        }
    }
}

// ---------------------------------------------------------------------------
extern "C" void kernel_launch(void* const* d_in, const int* in_sizes, int n_in,
                              void* d_out, int out_size, void* d_ws, size_t ws_size,
                              hipStream_t stream) {
    const float* x    = (const float*)d_in[0];
    const float* ctx  = (const float*)d_in[1];
    const float* Wq   = (const float*)d_in[2];
    const float* bq   = (const float*)d_in[3];
    const float* Wk   = (const float*)d_in[4];
    const float* bk   = (const float*)d_in[5];
    const float* Wv   = (const float*)d_in[6];
    const float* bv   = (const float*)d_in[7];
    const float* Wo   = (const float*)d_in[8];
    const float* bo   = (const float*)d_in[9];
    float* out = (float*)d_out;

    float* ktv  = (float*)d_ws;                 // B * 16*16
    float* ksum = ktv + B_ * DH * DH;           // B * 16
    const int nz = B_ * (DH * DH + DH);

    lca_zero_ws<<<(nz + 255) / 256, 256, 0, stream>>>(ktv, nz);

    dim3 grid1(N_ / 128, B_);
    lca_kv_reduce<<<grid1, 128, 0, stream>>>(ctx, Wk, bk, Wv, bv, ktv, ksum);

    dim3 grid2(M_ / 128, B_);
    lca_q_out<<<grid2, 128, 0, stream>>>(x, Wq, bq, ktv, ksum, Wo, bo, out);
}